// MultiSpectralPatchDCT2D_61942018343485
// MI455X (gfx1250) — compile-verified
//
#include <hip/hip_runtime.h>
#include <hip/hip_bf16.h>

// ---- types ----
typedef __attribute__((ext_vector_type(2))) float v2f;
typedef __attribute__((ext_vector_type(4))) float f4;
typedef __attribute__((ext_vector_type(8))) float v8f;

#define B_     16
#define C_     64
#define H_     224
#define W_     224
#define PH_    16
#define PW_    16
#define M_     16
#define HP_    (H_ / PH_)          // 14
#define WP_    (W_ / PW_)          // 14
#define NPB_   (HP_ * WP_)         // 196 patches per batch
#define NPTOT_ (B_ * NPB_)         // 3136 patches total
#define NTILE_ (NPTOT_ / 16)       // 196 wave-tiles of 16 patches
#define KDIM_  (PH_ * PW_)         // 256
#define HW4_   (H_ * W_ / 4)       // 12544 float4 per image plane

// =====================================================================
// Kernel 1: channel mean.  x[B,C,H,W] -> mean[B,H,W]   (205.5 MB -> 3.2 MB)
// One thread per float4 of mean; 64 strided NT loads, coalesced per wave.
// =====================================================================
__global__ __launch_bounds__(256) void k_mean(const f4* __restrict__ x,
                                              f4* __restrict__ mean) {
  int idx = blockIdx.x * 256 + threadIdx.x;        // 0 .. 200703 (exact grid)
  int b = idx / HW4_;
  int p = idx - b * HW4_;
  const f4* src = x + (size_t)b * C_ * HW4_ + p;
  float sx = 0.f, sy = 0.f, sz = 0.f, sw = 0.f;
#pragma unroll 4
  for (int c = 0; c < C_; ++c) {
    f4 v = __builtin_nontemporal_load(src + (size_t)c * HW4_);
    sx += v.x; sy += v.y; sz += v.z; sw += v.w;
  }
  const float s = 1.0f / (float)C_;
  f4 r; r.x = sx * s; r.y = sy * s; r.z = sz * s; r.w = sw * s;
  mean[idx] = r;
}

// =====================================================================
// Kernel 2: patch DCT via V_WMMA_F32_16X16X4_F32.
// One wave (32 lanes) per tile of 16 consecutive global patches:
//   D[16 patches x 16 bases] = A[16 x 256] * B[256 x 16],  K in chunks of 4.
// A/B staged in LDS with pitch 260 floats (bank-conflict-free b64 reads).
// =====================================================================
#define PITCH_ 260

__global__ __launch_bounds__(32) void k_dct(const float* __restrict__ mean,
                                            const float* __restrict__ bases,
                                            float* __restrict__ coeff) {
  __shared__ float ldsA[16 * PITCH_];   // 16 patches x 256 px (padded)
  __shared__ float ldsB[16 * PITCH_];   // 16 bases   x 256 px (padded)

  const int lane = threadIdx.x;         // 0..31
  const int tile = blockIdx.x;          // 0..195

  // ---- stage basis table (16 KB) into LDS, 16B-aligned b128 stores ----
#pragma unroll
  for (int n = 0; n < 16; ++n) {
#pragma unroll
    for (int it = 0; it < 2; ++it) {
      int t = it * 32 + lane;                                   // 0..63 float4
      f4 v = ((const f4*)(bases + n * KDIM_))[t];
      *(f4*)&ldsB[n * PITCH_ + t * 4] = v;
    }
  }

  // ---- stage 16 patches of the mean image into LDS ----
  for (int p = 0; p < 16; ++p) {
    int gp = tile * 16 + p;
    int b  = gp / NPB_;
    int r  = gp - b * NPB_;
    int hp = r / WP_;
    int wp = r - hp * WP_;
    const float* src = mean + ((size_t)(b * H_) + hp * PH_) * W_ + wp * PW_;
#pragma unroll
    for (int it = 0; it < 2; ++it) {
      int t  = it * 32 + lane;      // 0..63
      int i  = t >> 2;              // patch row 0..15
      int j4 = t & 3;               // float4 within row
      f4 v = *(const f4*)(src + i * W_ + j4 * 4);
      *(f4*)&ldsA[p * PITCH_ + i * PW_ + j4 * 4] = v;
    }
  }
  __syncthreads();

  // ---- WMMA accumulation: 64 x (16x16x4, f32) ----
  // ISA layouts (7.12.2): A lane(0-15)=M, holds K = 4k+{0,1}; lane(16-31) K = 4k+{2,3}.
  //                       B VGPR0 = rows K=4k (lanes 0-15) / K=4k+2 (lanes 16-31), VGPR1 = +1.
  const int half = lane >> 4;       // 0 or 1
  const int sub  = lane & 15;       // M for A, N for B
  v8f c = {0.f, 0.f, 0.f, 0.f, 0.f, 0.f, 0.f, 0.f};
  for (int kk = 0; kk < 64; ++kk) {
    int koff = kk * 4 + half * 2;
    v2f a = *(const v2f*)&ldsA[sub * PITCH_ + koff];
    v2f b = *(const v2f*)&ldsB[sub * PITCH_ + koff];
    c = __builtin_amdgcn_wmma_f32_16x16x4_f32(
        /*neg_a=*/false, a, /*neg_b=*/false, b,
        /*c_mod=*/(short)0, c, /*reuse_a=*/false, /*reuse_b=*/false);
  }

  // ---- D layout: VGPR r, lane -> coeff[patch = r + 8*half][basis = sub] ----
  // coeff stored as [B][M][Hp][Wp]
#pragma unroll
  for (int r = 0; r < 8; ++r) {
    int p  = r + 8 * half;
    int gp = tile * 16 + p;
    int b  = gp / NPB_;
    int rr = gp - b * NPB_;
    int hp = rr / WP_;
    int wp = rr - hp * WP_;
    coeff[(((size_t)b * M_ + sub) * HP_ + hp) * WP_ + wp] = c[r];
  }
}

// =====================================================================
// Kernel 3: nearest 16x16 upsample broadcast.  coeff[B,M,14,14] -> out[B,M,224,224]
// One thread per output float4 (each float4 is inside a single patch),
// non-temporal stores for the 51.4 MB write-once output.
// =====================================================================
__global__ __launch_bounds__(256) void k_upsample(const float* __restrict__ coeff,
                                                  f4* __restrict__ out) {
  int idx = blockIdx.x * 256 + threadIdx.x;   // 0 .. 3,211,263 (exact grid)
  int w4   = idx % (W_ / 4);                  // 0..55
  int rest = idx / (W_ / 4);
  int h    = rest % H_;
  int bm   = rest / H_;                       // b*16 + m
  int hp = h >> 4;
  int wp = (w4 * 4) >> 4;
  float v = coeff[((size_t)bm * HP_ + hp) * WP_ + wp];
  f4 o; o.x = v; o.y = v; o.z = v; o.w = v;
  __builtin_nontemporal_store(o, out + idx);
}

// =====================================================================
extern "C" void kernel_launch(void* const* d_in, const int* in_sizes, int n_in,
                              void* d_out, int out_size, void* d_ws, size_t ws_size,
                              hipStream_t stream) {
  const float* x     = (const float*)d_in[0];   // [16,64,224,224] fp32
  const float* bases = (const float*)d_in[1];   // [16,16,16] fp32
  float* out = (float*)d_out;                   // [16,16,224,224] fp32

  float* mean  = (float*)d_ws;                  // 802,816 floats (3.2 MB)
  float* coeff = mean + (size_t)B_ * H_ * W_;   // 50,176 floats  (0.2 MB)

  // 1) channel mean: 200,704 float4 threads
  {
    int n4 = B_ * H_ * W_ / 4;                  // 200704 = 784 * 256
    k_mean<<<n4 / 256, 256, 0, stream>>>((const f4*)x, (f4*)mean);
  }
  // 2) WMMA patch-DCT: 196 tiles, one wave each
  {
    k_dct<<<NTILE_, 32, 0, stream>>>(mean, bases, coeff);
  }
  // 3) broadcast upsample: 3,211,264 float4 threads
  {
    int n4 = B_ * M_ * H_ * W_ / 4;             // 3,211,264 = 12544 * 256
    k_upsample<<<n4 / 256, 256, 0, stream>>>(coeff, (f4*)out);
  }
}